// MatrixHyperlayer_84181359001735
// MI455X (gfx1250) — compile-verified
//
#include <hip/hip_runtime.h>
#include <math.h>

// ---------------------------------------------------------------------------
// MatrixHyperlayer scatter for MI455X (gfx1250, wave32).
//
// Per k (one wave per k): 16 (i,j,val) pairs; out[i,:] += val * x[j,:] (128 f32).
// Atomic-bound: L2 f32 atomics dominate; x & out are L2-resident (192MB L2).
//
// diag(val16) x X_gathered(16x128) is computed with V_WMMA_F32_16X16X4_F32
// (bit-exact: zero-padded diagonal A => each D element is val_m*x + exact 0s).
// Feature columns are mapped to tiles INTERLEAVED: tile t holds columns
// {8n+t}, so each lane's B data per gathered row is 8 consecutive floats
// (two b128 loads) and each lane's 8 atomic targets per output row are
// consecutive (one base address + immediate offsets).
//
// Pair->row remap for atomic reduction: the 4 floor/ceil neighbors have only
// 2 distinct output rows (i = floor(mean0) for masks (T,*), ceil for (F,*)).
// Placing them at rows {0,1} and {8,9} lets acc[0]+acc[1] pre-merge the two
// scaled rows in registers for BOTH lane halves: 56 instead of 64 atomics/k.
// ---------------------------------------------------------------------------

#define IN_NUM      50000
#define OUT_NUM     50000
#define K_EDGES     131072
#define ADDITIONAL  12
#define FEAT        128
#define WAVES_PER_BLOCK 8
#define EPS_F       1e-7f

typedef __attribute__((ext_vector_type(2))) float v2f;
typedef __attribute__((ext_vector_type(8))) float v8f;

__device__ __forceinline__ float sigmoid_f(float z) {
    return 1.0f / (1.0f + expf(-z));
}

__global__ __launch_bounds__(256) void hyper_scatter_kernel(
    const float* __restrict__ params,      // (K,4)
    const float* __restrict__ x,           // (IN_NUM, 128)
    const int*   __restrict__ sampled,     // (K,12,2) : [:,:,0]=i(out), [:,:,1]=j(in)
    float*       __restrict__ out)         // (OUT_NUM, 128)
{
    const int lane = threadIdx.x & 31;
    const int wave = threadIdx.x >> 5;
    // force wave-uniform k into an SGPR so params load becomes s_load_b128
    const int k  = __builtin_amdgcn_readfirstlane(blockIdx.x * WAVES_PER_BLOCK + wave);
    const int lo = lane & 15;   // matrix row M (duplicated in both halves)
    const int hi = lane >> 4;

    // ---- per-k hyper params (scalar) ----
    const float4 p = ((const float4*)params)[k];
    const float mean0 = sigmoid_f(p.x) * (float)(OUT_NUM - 1);
    const float mean1 = sigmoid_f(p.y) * (float)(IN_NUM - 1);
    const float z     = p.z + 2.0f;                                   // SIGMA_BOOST
    const float sp    = fmaxf(z, 0.0f) + log1pf(expf(-fabsf(z)));     // softplus
    const float sigma = (sp + EPS_F) * ((float)OUT_NUM * 0.2f);       // *size*SIGMA_SCALE
    const float inv_std = rsqrtf(sigma + EPS_F);                      // sqrt(1/(eps+sigma))
    const float value = sigmoid_f(p.w);

    // ---- pair assigned to row M = lo -------------------------------------
    // M=0:(T,T)  M=1:(T,F)  M=8:(F,T)  M=9:(F,F)   (True -> floor)
    // M=2..7  : sampled 0..5      M=10..15 : sampled 6..11
    const bool is_nb = (lo == 0) | (lo == 1) | (lo == 8) | (lo == 9);
    int   ii, jj;
    float fi, fj;
    if (is_nb) {
        const float f0 = floorf(mean0), c0 = ceilf(mean0);
        const float f1 = floorf(mean1), c1 = ceilf(mean1);
        fi = (lo & 8) ? c0 : f0;            // rows 0,1 -> floor ; 8,9 -> ceil
        fj = (lo & 1) ? c1 : f1;
        ii = (int)fi;  jj = (int)fj;
    } else {
        const int a = (lo < 8) ? (lo - 2) : (lo - 4);   // sampled index 0..11
        const int2 sij = ((const int2*)sampled)[k * ADDITIONAL + a];
        ii = sij.x;  jj = sij.y;
        fi = (float)ii;  fj = (float)jj;
    }

    const float d0 = (fi - mean0) * inv_std;
    const float d1 = (fj - mean1) * inv_std;
    const float prop = expf(-0.5f * (d0 * d0 + d1 * d1));

    // sum over the 16 pairs (halves hold identical data -> 16-wide xor tree)
    float s = prop;
    s += __shfl_xor(s, 1);
    s += __shfl_xor(s, 2);
    s += __shfl_xor(s, 4);
    s += __shfl_xor(s, 8);
    const float denom = s + 16.0f * EPS_F;      // sum(props + eps)
    const float myval = value * prop / denom;   // this lane's pair weight

    // ---- WMMA A operands --------------------------------------------------
    // A_c (16x4 f32, 2 VGPRs): lane l holds M=l&15, VGPR r holds K=2*(l>>4)+r.
    // diag slice: A_c[m,kk] = val_m iff m == 4c+kk.
    v2f A[4];
#pragma unroll
    for (int c = 0; c < 4; ++c) {
        A[c].x = (lo == 4 * c + 2 * hi + 0) ? myval : 0.0f;
        A[c].y = (lo == 4 * c + 2 * hi + 1) ? myval : 0.0f;
    }

    // ---- B preload --------------------------------------------------------
    // B_c VGPR r, this lane = rows of pair q = 4c + 2*hi + r.
    // Tile t <-> feature column 8*lo + t, so this lane needs
    // x[j_q][8*lo .. 8*lo+7]: two b128 loads per (c,r).
    float xs[8][8];                 // [2c+r][t]
#pragma unroll
    for (int c = 0; c < 4; ++c) {
#pragma unroll
        for (int r = 0; r < 2; ++r) {
            const int jq = __shfl(jj, 4 * c + 2 * hi + r);
            const float4* rowp = (const float4*)(x + jq * FEAT + 8 * lo);
            const float4 a4 = rowp[0];
            const float4 b4 = rowp[1];
            float* d = xs[2 * c + r];
            d[0] = a4.x; d[1] = a4.y; d[2] = a4.z; d[3] = a4.w;
            d[4] = b4.x; d[5] = b4.y; d[6] = b4.z; d[7] = b4.w;
        }
    }

    // ---- scatter bases ----------------------------------------------------
    // D tile t, VGPR r: element (M = r + 8*hi, N = lo) -> out[i_M*128 + 8*lo + t].
    // VGPR 0 and 1 target the same output row (floor / ceil of mean0 in the
    // low / high halves respectively), so they are pre-merged: no dstb[1].
    float* dstb[8];
#pragma unroll
    for (int r = 0; r < 8; ++r) {
        if (r == 1) continue;
        const int iq = __shfl(ii, r + 8 * hi);
        dstb[r] = out + iq * FEAT + 8 * lo;
    }

    // ---- 8 interleaved column tiles ---------------------------------------
#pragma unroll
    for (int t = 0; t < 8; ++t) {
        v8f acc = {0.0f, 0.0f, 0.0f, 0.0f, 0.0f, 0.0f, 0.0f, 0.0f};
#pragma unroll
        for (int c = 0; c < 4; ++c) {
            v2f B;
            B.x = xs[2 * c + 0][t];
            B.y = xs[2 * c + 1][t];
            acc = __builtin_amdgcn_wmma_f32_16x16x4_f32(
                false, A[c], false, B, (short)0, acc, false, false);
        }
        // rows 0+1 (low: floor-pair, high: rows 8,9 = ceil-pair) share i
        __hip_atomic_fetch_add(dstb[0] + t, acc[0] + acc[1], __ATOMIC_RELAXED,
                               __HIP_MEMORY_SCOPE_AGENT);
#pragma unroll
        for (int r = 2; r < 8; ++r) {
            __hip_atomic_fetch_add(dstb[r] + t, acc[r], __ATOMIC_RELAXED,
                                   __HIP_MEMORY_SCOPE_AGENT);
        }
    }
}

extern "C" void kernel_launch(void* const* d_in, const int* in_sizes, int n_in,
                              void* d_out, int out_size, void* d_ws, size_t ws_size,
                              hipStream_t stream) {
    const float* params  = (const float*)d_in[0];   // (K,4) f32
    const float* x       = (const float*)d_in[1];   // (50000,128) f32
    const int*   sampled = (const int*)d_in[2];     // (K,12,2) i32
    float*       out     = (float*)d_out;           // (50000,128) f32

    // harness poisons d_out; we accumulate with atomics, so zero it first
    hipMemsetAsync(out, 0, (size_t)out_size * sizeof(float), stream);

    dim3 grid(K_EDGES / WAVES_PER_BLOCK);
    dim3 block(32 * WAVES_PER_BLOCK);
    hipLaunchKernelGGL(hyper_scatter_kernel, grid, block, 0, stream,
                       params, x, sampled, out);
}